// TestModule_48954037240021
// MI455X (gfx1250) — compile-verified
//
#include <hip/hip_runtime.h>
#include <stdint.h>

typedef __attribute__((ext_vector_type(2))) float v2f;
typedef __attribute__((ext_vector_type(8))) float v8f;

#define QMAXF 15.0f
#define EPSF  1e-8f

// Order-preserving float <-> uint key (unsigned compare == float total order).
__device__ __forceinline__ unsigned f2key(float f) {
    unsigned u = __float_as_uint(f);
    return (u & 0x80000000u) ? ~u : (u | 0x80000000u);
}
__device__ __forceinline__ float key2f(unsigned k) {
    unsigned u = (k & 0x80000000u) ? (k ^ 0x80000000u) : ~k;
    return __uint_as_float(u);
}

__device__ __forceinline__ float fq(float v, float s, float zp) {
    // matches jnp: clip(round(v/s)+zp, 0, 15); rintf = round-half-even like jnp.round
    float q = fminf(fmaxf(rintf(v / s) + zp, 0.0f), QMAXF);
    return (q - zp) * s;
}

// ws layout (as uint/float): [0]=min key, [1]=max key, [2]=scale_x, [3]=zp_x, [4..53]=Wq (10x5)
__global__ void init_ws_kernel(unsigned* __restrict__ ws) {
    ws[0] = 0xFFFFFFFFu;  // running unsigned-min of keys
    ws[1] = 0x00000000u;  // running unsigned-max of keys
}

// Pass 1: global min/max of x (160 MB, RT loads -> leaves x resident in 192MB L2)
__global__ void __launch_bounds__(256) minmax_kernel(const float4* __restrict__ x4,
                                                     long n4, long n,
                                                     unsigned* __restrict__ ws) {
    long i = (long)blockIdx.x * blockDim.x + threadIdx.x;
    const long stride = (long)gridDim.x * blockDim.x;
    float vmin = INFINITY, vmax = -INFINITY;
    for (long j = i; j < n4; j += stride) {
        float4 v = x4[j];  // global_load_b128
        vmin = fminf(vmin, fminf(fminf(v.x, v.y), fminf(v.z, v.w)));
        vmax = fmaxf(vmax, fmaxf(fmaxf(v.x, v.y), fmaxf(v.z, v.w)));
    }
    // scalar tail (n % 4), folded into thread 0 before the wave reduction
    if (blockIdx.x == 0 && threadIdx.x == 0) {
        const float* xs = (const float*)x4;
        for (long j = n4 * 4; j < n; ++j) {
            vmin = fminf(vmin, xs[j]);
            vmax = fmaxf(vmax, xs[j]);
        }
    }
    // wave32 butterfly reduction
    #pragma unroll
    for (int m = 16; m > 0; m >>= 1) {
        vmin = fminf(vmin, __shfl_xor(vmin, m, 32));
        vmax = fmaxf(vmax, __shfl_xor(vmax, m, 32));
    }
    if ((threadIdx.x & 31) == 0) {
        atomicMin(&ws[0], f2key(vmin));
        atomicMax(&ws[1], f2key(vmax));
    }
}

// Pass 2 prep: x quant params + fake-quantized W (tiny, 1 thread)
__global__ void prep_kernel(const float* __restrict__ W, unsigned* __restrict__ wsu) {
    float* wsf = (float*)wsu;
    float xmin = key2f(wsu[0]);
    float xmax = key2f(wsu[1]);
    float sx = fmaxf((xmax - xmin) / QMAXF, EPSF);
    float zx = rintf(-xmin / sx);
    wsf[2] = sx;
    wsf[3] = zx;
    float wmin = INFINITY, wmax = -INFINITY;
    for (int i = 0; i < 50; ++i) {
        wmin = fminf(wmin, W[i]);
        wmax = fmaxf(wmax, W[i]);
    }
    float sw = fmaxf((wmax - wmin) / QMAXF, EPSF);
    float zw = rintf(-wmin / sw);
    for (int i = 0; i < 50; ++i) {
        float q = fminf(fmaxf(rintf(W[i] / sw) + zw, 0.0f), QMAXF);
        wsf[4 + i] = (q - zw) * sw;
    }
}

// Pass 3: fake-quant x + linear via V_WMMA_F32_16X16X4_F32.
// Each wave handles 16 rows of x per tile:
//   A (16x4 f32): lane half g, VGPR j -> K=2g+j, M=lane%16
//   B (4x16 f32): lane half g, VGPR j -> K=2g+j, N=lane%16 (Wq^T, cols>=10 zero)
//   C/D (16x16):  N=lane%16, M=vgpr+8g ; C pre-loaded with bias broadcast
// K=5 split across two WMMAs: K0..3, then K4 zero-padded.
__global__ void __launch_bounds__(256) qlinear_wmma_kernel(
        const float* __restrict__ x, const float* __restrict__ bias,
        const unsigned* __restrict__ wsu, float* __restrict__ out,
        long nrows, long ntiles) {
    const float* wsf = (const float*)wsu;
    const float sx = wsf[2];
    const float zx = wsf[3];

    const int lane = threadIdx.x & 31;
    const int g    = lane >> 4;   // lane half
    const int mn   = lane & 15;   // M for A, N for B/C
    const bool activeN = (mn < 10);
    const int k0 = 2 * g;

    // Loop-invariant B operands (L2-resident tiny tables)
    v2f B1, B2;
    B1.x = activeN ? wsf[4 + mn * 5 + k0]     : 0.0f;
    B1.y = activeN ? wsf[4 + mn * 5 + k0 + 1] : 0.0f;
    B2.x = (activeN && g == 0) ? wsf[4 + mn * 5 + 4] : 0.0f;  // K=4 row only
    B2.y = 0.0f;
    const float bval = activeN ? bias[mn] : 0.0f;

    const long wave   = (long)blockIdx.x * (blockDim.x >> 5) + (threadIdx.x >> 5);
    const long nwaves = (long)gridDim.x * (blockDim.x >> 5);

    for (long t = wave; t < ntiles; t += nwaves) {
        const long row0 = t << 4;
        const bool full = (row0 + 16 <= nrows);       // wave-uniform

        long r = row0 + mn;
        if (r >= nrows) r = nrows - 1;                // clamp keeps EXEC full for WMMA
        const float* xr = x + r * 5;

        // Load + fake-quant A tile (x should hit L2 after pass 1)
        float a0 = fq(xr[k0],     sx, zx);
        float a1 = fq(xr[k0 + 1], sx, zx);
        float a4 = fq(xr[4],      sx, zx);

        v2f A1; A1.x = a0; A1.y = a1;
        v2f A2; A2.x = (g == 0) ? a4 : 0.0f; A2.y = 0.0f;

        v8f c = {bval, bval, bval, bval, bval, bval, bval, bval};
        c = __builtin_amdgcn_wmma_f32_16x16x4_f32(false, A1, false, B1,
                                                  (short)0, c, false, false);
        c = __builtin_amdgcn_wmma_f32_16x16x4_f32(false, A2, false, B2,
                                                  (short)0, c, false, false);

        // Store D rows M=v+8g at column mn. Fast path: one base pointer, 8 NT
        // stores with immediate offsets (40B apart) -> no per-store exec/branch.
        float* base = out + row0 * 10 + (long)g * 80 + mn;
        if (full & activeN) {
            #pragma unroll
            for (int v = 0; v < 8; ++v) {
                __builtin_nontemporal_store(c[v], base + v * 10);
            }
        } else if (activeN) {
            // tail tile only (never taken when nrows % 16 == 0)
            #pragma unroll
            for (int v = 0; v < 8; ++v) {
                if (row0 + 8 * g + v < nrows) {
                    __builtin_nontemporal_store(c[v], base + v * 10);
                }
            }
        }
    }
}

extern "C" void kernel_launch(void* const* d_in, const int* in_sizes, int n_in,
                              void* d_out, int out_size, void* d_ws, size_t ws_size,
                              hipStream_t stream) {
    const float* x = (const float*)d_in[0];   // [B,5]
    const float* W = (const float*)d_in[1];   // [10,5]
    const float* b = (const float*)d_in[2];   // [10]
    float* out = (float*)d_out;               // [B,10]
    unsigned* ws = (unsigned*)d_ws;

    const long n     = (long)in_sizes[0];     // B*5
    const long nrows = n / 5;                 // B
    const long ntiles = (nrows + 15) / 16;
    const long n4 = n / 4;

    init_ws_kernel<<<1, 1, 0, stream>>>(ws);
    minmax_kernel<<<4096, 256, 0, stream>>>((const float4*)x, n4, n, ws);
    prep_kernel<<<1, 1, 0, stream>>>(W, ws);
    qlinear_wmma_kernel<<<8192, 256, 0, stream>>>(x, b, ws, out, nrows, ntiles);
}